// DistillationLoss_Centroid_13039520710819
// MI455X (gfx1250) — compile-verified
//
#include <hip/hip_runtime.h>

// Native vector types (clang ext_vector so __builtin_nontemporal_load works)
typedef float vf4 __attribute__((ext_vector_type(4)));
typedef float v2f __attribute__((ext_vector_type(2)));
typedef float v8f __attribute__((ext_vector_type(8)));

#define NBLK 1024
#define TPB  256
#define WPB  (TPB / 32)   // waves per block (wave32)
#define DIM  512          // reference embedding dim (4 float4 chunks per lane)

// ds_swizzle group-of-32 xor patterns: imm = (xor_mask<<10) | (or<<5) | and_mask
#define SWZ_X8 0x201F
#define SWZ_X4 0x101F
#define SWZ_X2 0x081F
#define SWZ_X1 0x041F

static __device__ __forceinline__ float swz_add(float x, int imm) {
    int xi = __builtin_bit_cast(int, x);
    int yi;
    switch (imm) {  // imm must be a literal constant for the builtin
        case SWZ_X8: yi = __builtin_amdgcn_ds_swizzle(xi, SWZ_X8); break;
        case SWZ_X4: yi = __builtin_amdgcn_ds_swizzle(xi, SWZ_X4); break;
        case SWZ_X2: yi = __builtin_amdgcn_ds_swizzle(xi, SWZ_X2); break;
        default:     yi = __builtin_amdgcn_ds_swizzle(xi, SWZ_X1); break;
    }
    return x + __builtin_bit_cast(float, yi);
}

// ---------------------------------------------------------------------------
// Kernel 1: masked streaming SSE. One wave per sample (stride loop).
// Rows with label >= num_old are skipped entirely (halves HBM traffic).
// All 8 b128 loads for a row are issued before any compute (max loads in
// flight); the next masked row is prefetched while the current one computes.
// Per-lane running accumulator; final 32->1 reduce via WMMA f32 16x16x4
// (A = ones => D = column-sums of B, independent of B's 4x16 lane mapping)
// followed by 4 immediate ds_swizzle xor-adds. Deterministic: no float atomics.
// ---------------------------------------------------------------------------
__global__ __launch_bounds__(TPB) void dist_loss_partial(
    const float* __restrict__ emb,      // [batch, DIM]
    const float* __restrict__ cent,     // [nclass, DIM]
    const int*   __restrict__ labels,   // [batch]
    const int*   __restrict__ num_old_p,// [1]
    float* __restrict__ ws_sum,         // [NBLK] partial SSE sums
    int*   __restrict__ ws_cnt,         // [NBLK] partial mask counts
    int batch)
{
    const int lane    = threadIdx.x & 31;
    const int wid     = threadIdx.x >> 5;
    const int gwave   = blockIdx.x * WPB + wid;
    const int nwave   = gridDim.x * WPB;
    const int num_old = num_old_p[0];

    float acc = 0.0f;   // per-lane running sum of squared diffs (masked rows only)
    int   cnt = 0;      // per-wave masked-sample count (wave-uniform)

    for (int s = gwave; s < batch; s += nwave) {
        const int lab = labels[s];                 // uniform across the wave
        if (lab < num_old) {
            cnt++;
            const vf4* e4 = (const vf4*)(emb  + (size_t)s   * DIM) + lane;
            const vf4* c4 = (const vf4*)(cent + (size_t)lab * DIM) + lane;

            // Issue all loads up front: 4 NT embedding chunks + 4 centroid chunks
            vf4 ev[4], cv[4];
            #pragma unroll
            for (int j = 0; j < 4; ++j) ev[j] = __builtin_nontemporal_load(&e4[j * 32]);
            #pragma unroll
            for (int j = 0; j < 4; ++j) cv[j] = c4[j * 32];

            // Prefetch the next masked row (each lane covers a 64B slice ->
            // 32 lanes span the full 2KB row) while this row's loads land.
            const int s_next = s + nwave;
            if (s_next < batch) {
                if (labels[s_next] < num_old) {
                    __builtin_prefetch(emb + (size_t)s_next * DIM + lane * 16, 0, 1);
                }
            }

            #pragma unroll
            for (int j = 0; j < 4; ++j) {
                float dx = ev[j].x - cv[j].x;
                float dy = ev[j].y - cv[j].y;
                float dz = ev[j].z - cv[j].z;
                float dw = ev[j].w - cv[j].w;
                acc = fmaf(dx, dx, acc);
                acc = fmaf(dy, dy, acc);
                acc = fmaf(dz, dz, acc);
                acc = fmaf(dw, dw, acc);
            }
        }
    }

    // ---- wave reduction: 32 lanes -> 1 -------------------------------------
    // WMMA: D = A(ones,16x4) x B + 0. With B-VGPR0 = acc, B-VGPR1 = 0, every
    // lane's acc occupies exactly one (k,j) slot of B, so D[i][j] = colsum_j
    // and D's VGPR0 holds the 16 column sums in lanes 0-15 (mirrored 16-31).
    // All 32 lanes reach here unconditionally -> EXEC is all ones as required.
    v2f a; a[0] = 1.0f; a[1] = 1.0f;
    v2f b; b[0] = acc;  b[1] = 0.0f;
    v8f c = {};
    c = __builtin_amdgcn_wmma_f32_16x16x4_f32(
            /*neg_a=*/false, a, /*neg_b=*/false, b,
            /*c_mod=*/(short)0, c, /*reuse_a=*/false, /*reuse_b=*/false);
    float colsum = c[0];
    colsum = swz_add(colsum, SWZ_X8);   // xor stays within each 16-lane half
    colsum = swz_add(colsum, SWZ_X4);
    colsum = swz_add(colsum, SWZ_X2);
    colsum = swz_add(colsum, SWZ_X1);   // lane 0 = sum of lanes 0..15 colsums

    __shared__ float wsum[WPB];
    __shared__ int   wcnt[WPB];
    if (lane == 0) { wsum[wid] = colsum; wcnt[wid] = cnt; }
    __syncthreads();

    if (threadIdx.x == 0) {
        float s = 0.0f; int n = 0;
        #pragma unroll
        for (int i = 0; i < WPB; ++i) { s += wsum[i]; n += wcnt[i]; }  // fixed order
        ws_sum[blockIdx.x] = s;
        ws_cnt[blockIdx.x] = n;
    }
}

// ---------------------------------------------------------------------------
// Kernel 2: deterministic tree-reduce of NBLK partials, emit scalar loss.
// loss = (count == 0) ? total : total / count, total = (sum SSE) / DIM
// ---------------------------------------------------------------------------
__global__ __launch_bounds__(TPB) void dist_loss_final(
    const float* __restrict__ ws_sum,
    const int*   __restrict__ ws_cnt,
    float* __restrict__ out, int nblk, float inv_dim)
{
    __shared__ float ssum[TPB];
    __shared__ int   scnt[TPB];
    float s = 0.0f; int n = 0;
    for (int i = threadIdx.x; i < nblk; i += TPB) { s += ws_sum[i]; n += ws_cnt[i]; }
    ssum[threadIdx.x] = s;
    scnt[threadIdx.x] = n;
    __syncthreads();
    for (int off = TPB / 2; off > 0; off >>= 1) {
        if (threadIdx.x < off) {
            ssum[threadIdx.x] += ssum[threadIdx.x + off];   // fixed pairing
            scnt[threadIdx.x] += scnt[threadIdx.x + off];
        }
        __syncthreads();
    }
    if (threadIdx.x == 0) {
        float total = ssum[0] * inv_dim;
        int   count = scnt[0];
        out[0] = (count == 0) ? total : (total / (float)count);
    }
}

extern "C" void kernel_launch(void* const* d_in, const int* in_sizes, int n_in,
                              void* d_out, int out_size, void* d_ws, size_t ws_size,
                              hipStream_t stream) {
    const float* emb      = (const float*)d_in[0];   // embeddings [B, D] f32
    const float* cent     = (const float*)d_in[1];   // centroids  [C, D] f32
    const int*   labels   = (const int*)d_in[2];     // labels [B] i32
    const int*   num_old  = (const int*)d_in[3];     // scalar i32

    const int batch = in_sizes[2];

    float* ws_sum = (float*)d_ws;
    int*   ws_cnt = (int*)((float*)d_ws + NBLK);

    dist_loss_partial<<<NBLK, TPB, 0, stream>>>(emb, cent, labels, num_old,
                                                ws_sum, ws_cnt, batch);
    dist_loss_final<<<1, TPB, 0, stream>>>(ws_sum, ws_cnt, (float*)d_out,
                                           NBLK, 1.0f / (float)DIM);
}